// RepeatEncoder_87806311400084
// MI455X (gfx1250) — compile-verified
//
#include <hip/hip_runtime.h>
#include <stdint.h>

// Problem constants (from the reference): x is (B=32, L=720, C=862) f32,
// output is spikes (T=4, B=32, C=862, L=720) f32.
#define T_STEPS 4
#define BATCH   32
#define LDIM    720
#define CDIM    862
#define TILE    32
#define ROWS    8          // blockDim.y; TILE/ROWS = 4 rows per thread per phase

// LIF (tau=2, v_reset=0, v_th=1, decay_input=True, hard reset):
//   v <- (v + x) / 2 ; s = (v >= 1) ; v <- s ? 0 : v
// Spikes depend only on x, so compute all 4 time steps in registers and fan
// out 4 transposed, coalesced, non-temporal stores.

__global__ __launch_bounds__(256)
void lif_repeat_transpose(const float* __restrict__ x, float* __restrict__ out) {
    // +1 padding: transposed column reads hit distinct banks (33 coprime w/ 64)
    __shared__ float tile[TILE][TILE + 1];

    const int tx = threadIdx.x;           // 0..31
    const int ty = threadIdx.y;           // 0..7
    const int l0 = blockIdx.x * TILE;     // tile origin along L
    const int c0 = blockIdx.y * TILE;     // tile origin along C
    const int b  = blockIdx.z;            // batch

    const uint64_t src =
        (uint64_t)(uintptr_t)(x + (size_t)b * ((size_t)LDIM * CDIM));

    // ---- Phase 1: async DMA global -> LDS, coalesced over C ----------------
    // Boundary tiles: clamp addresses (keeps EXEC all-1s; clamped duplicates
    // are never stored). B32 transfers => always naturally aligned even though
    // C=862 leaves rows only 8B-aligned.
    #pragma unroll
    for (int r = 0; r < TILE / ROWS; ++r) {
        const int ll = ty + ROWS * r;                       // local row (L)
        int l = l0 + ll; if (l > LDIM - 1) l = LDIM - 1;
        int c = c0 + tx; if (c > CDIM - 1) c = CDIM - 1;
        const uint32_t goff = (uint32_t)(((uint32_t)l * CDIM + (uint32_t)c) * 4u);
        const uint32_t loff = (uint32_t)(uintptr_t)&tile[ll][tx]; // LDS byte addr
        // GVS mode: 64-bit SGPR base + 32-bit per-lane byte offset, dest VGPR
        // holds the LDS byte address. Tracked by ASYNCcnt.
        asm volatile("global_load_async_to_lds_b32 %0, %1, %2"
                     :: "v"(loff), "v"(goff), "s"(src)
                     : "memory");
    }
    // Wait for this wave's async transfers, then sync all waves in the block.
    asm volatile("s_wait_asynccnt 0x0" ::: "memory");
    __syncthreads();

    // ---- Phase 2: transposed read from LDS, LIF in registers, NT stores ----
    const int  l   = l0 + tx;             // now tx runs along L -> coalesced out
    const bool lok = (l < LDIM);

    #pragma unroll
    for (int r = 0; r < TILE / ROWS; ++r) {
        const int cl = ty + ROWS * r;     // local row (C)
        const int c  = c0 + cl;
        const float xv = tile[tx][cl];    // transposed, conflict-free (stride 33)

        float v = 0.0f;
        float s[T_STEPS];
        #pragma unroll
        for (int t = 0; t < T_STEPS; ++t) {
            v = 0.5f * (v + xv);
            const bool fire = (v >= 1.0f);
            s[t] = fire ? 1.0f : 0.0f;
            v    = fire ? 0.0f : v;       // hard reset to v_reset = 0
        }

        if (lok && c < CDIM) {
            #pragma unroll
            for (int t = 0; t < T_STEPS; ++t) {
                const size_t idx =
                    (((size_t)t * BATCH + (size_t)b) * CDIM + (size_t)c) * LDIM
                    + (size_t)l;
                __builtin_nontemporal_store(s[t], out + idx);
            }
        }
    }
}

extern "C" void kernel_launch(void* const* d_in, const int* in_sizes, int n_in,
                              void* d_out, int out_size, void* d_ws, size_t ws_size,
                              hipStream_t stream) {
    (void)in_sizes; (void)n_in; (void)out_size; (void)d_ws; (void)ws_size;
    const float* x   = (const float*)d_in[0];
    float*       out = (float*)d_out;

    dim3 block(TILE, ROWS, 1);                                  // 256 = 8 waves
    dim3 grid((LDIM + TILE - 1) / TILE,                         // 23
              (CDIM + TILE - 1) / TILE,                         // 27
              BATCH);                                           // 32
    hipLaunchKernelGGL(lif_repeat_transpose, grid, block, 0, stream, x, out);
}